// ISPPipeline_7181185319102
// MI455X (gfx1250) — compile-verified
//
#include <hip/hip_runtime.h>
#include <cstdint>
#include <cstddef>

// ---------------- problem geometry ----------------
#define HH 3072
#define WW 4096
#define H2 1536
#define W2 2048

// ---------------- CDNA5 feature probes ----------------
#if defined(__gfx1250__) && __has_builtin(__builtin_amdgcn_wmma_f32_16x16x4_f32)
#define USE_WMMA 1
#else
#define USE_WMMA 0
#endif

#if defined(__gfx1250__) && __has_builtin(__builtin_amdgcn_global_load_async_to_lds_b32)
#define USE_ASYNC 1
#else
#define USE_ASYNC 0
#endif

typedef __attribute__((ext_vector_type(2))) float v2f;
typedef __attribute__((ext_vector_type(8))) float v8f;

__device__ __forceinline__ float clip01(float v) { return fminf(fmaxf(v, 0.0f), 1.0f); }

// decompand (knee [0,.25,1]->[0,.5,1]) + black level 0.0625
__device__ __forceinline__ float decompand_bl(float x) {
    float y = (x <= 0.25f) ? (2.0f * x) : (0.5f + (x - 0.25f) * (2.0f / 3.0f));
    return clip01((y - 0.0625f) * (1.0f / 0.9375f));
}

// truncated-window count along one axis, radius r
__device__ __forceinline__ float wcount(int g, int n, int r) {
    int lo = max(g - r, 0), hi = min(g + r, n - 1);
    return (float)(hi - lo + 1);
}

// ---- tile loader: global -> LDS with zero padding; async on CDNA5 ----
__device__ __forceinline__ void tile_load(float* sm, int pitch,
                                          const float* __restrict__ g,
                                          int gy0, int gx0, int th, int tw,
                                          int gh, int gw, int tid, int nthr) {
    for (int i = tid; i < th * tw; i += nthr) {
        int r = i / tw, c = i % tw;
        int gy = gy0 + r, gx = gx0 + c;
        bool ok = (gy >= 0) & (gy < gh) & (gx >= 0) & (gx < gw);
#if USE_ASYNC
        if (ok) {
            __builtin_amdgcn_global_load_async_to_lds_b32(
                (__attribute__((address_space(1))) int*)(g + (size_t)gy * gw + gx),
                (__attribute__((address_space(3))) int*)(sm + r * pitch + c),
                0, 0);
        } else {
            sm[r * pitch + c] = 0.0f;
        }
#else
        sm[r * pitch + c] = ok ? g[(size_t)gy * gw + gx] : 0.0f;
#endif
    }
#if USE_ASYNC
    asm volatile("s_wait_asynccnt 0" ::: "memory");
#endif
}

// =============== K1: decompand + black level, split RGGB planes ===============
__global__ __launch_bounds__(256) void k_decompand(const float* __restrict__ x,
                                                   float* __restrict__ P) {
    int idx = blockIdx.x * 256 + threadIdx.x;     // one thread per 2x2 quad
    int y2 = idx / W2, x2 = idx % W2;
    const float2* r0 = (const float2*)(x + (size_t)(2 * y2) * WW);
    const float2* r1 = (const float2*)(x + (size_t)(2 * y2 + 1) * WW);
    float2 a = r0[x2], b = r1[x2];
    int o = y2 * W2 + x2;
    P[0 * H2 * W2 + o] = decompand_bl(a.x);
    P[1 * H2 * W2 + o] = decompand_bl(a.y);
    P[2 * H2 * W2 + o] = decompand_bl(b.x);
    P[3 * H2 * W2 + o] = decompand_bl(b.y);
}

// =============== K2: fused guided-filter denoise (r=2, eps=100) per plane ===============
// 32x32 output tile; all intermediates (m, var, a, b and their box means) stay in LDS.
#define NBLK_X (W2 / 32)
#define NBLK_Y (H2 / 32)
__global__ __launch_bounds__(256) void k_denoise(const float* __restrict__ P,
                                                 float* __restrict__ PDN,
                                                 float* __restrict__ BSUM) {
    const int plane = blockIdx.z;
    const int oy = blockIdx.y * 32, ox = blockIdx.x * 32;
    const int tid = threadIdx.x;
    const float* g = P + (size_t)plane * H2 * W2;

    __shared__ float Tin[40 * 42];
    __shared__ float Hs[40 * 36];
    __shared__ float Hs2[40 * 36];
    __shared__ float Aa[36 * 36];
    __shared__ float Bb[36 * 36];
    __shared__ float HA[36 * 32];
    __shared__ float HB[36 * 32];
    __shared__ float red[8];

    tile_load(Tin, 42, g, oy - 4, ox - 4, 40, 40, H2, W2, tid, 256);
    __syncthreads();

    // horizontal 5-tap sums of I and I^2
    for (int i = tid; i < 40 * 36; i += 256) {
        int r = i / 36, j = i % 36;
        const float* t = &Tin[r * 42 + j];
        float s = 0.f, s2 = 0.f;
#pragma unroll
        for (int k = 0; k < 5; k++) { float v = t[k]; s += v; s2 += v * v; }
        Hs[i] = s; Hs2[i] = s2;
    }
    __syncthreads();

    // vertical 5-tap -> m, var -> a, b (on 36x36 halo region)
    for (int i = tid; i < 36 * 36; i += 256) {
        int r = i / 36, j = i % 36;
        float s = 0.f, s2 = 0.f;
#pragma unroll
        for (int k = 0; k < 5; k++) { s += Hs[(r + k) * 36 + j]; s2 += Hs2[(r + k) * 36 + j]; }
        int gy = oy - 2 + r, gx = ox - 2 + j;
        float inv = 1.0f / (wcount(gy, H2, 2) * wcount(gx, W2, 2));
        float m = s * inv, m2 = s2 * inv;
        float var = m2 - m * m;
        float a = var / (var + 100.0f);
        Aa[i] = a; Bb[i] = m * (1.0f - a);
    }
    __syncthreads();

    // horizontal 5-tap of a, b
    for (int i = tid; i < 36 * 32; i += 256) {
        int r = i / 32, j = i % 32;
        float sa = 0.f, sb = 0.f;
#pragma unroll
        for (int k = 0; k < 5; k++) { sa += Aa[r * 36 + j + k]; sb += Bb[r * 36 + j + k]; }
        HA[i] = sa; HB[i] = sb;
    }
    __syncthreads();

    // vertical 5-tap -> q = mean(a)*I + mean(b); accumulate plane sum for AWB
    float lsum = 0.0f;
    for (int i = tid; i < 32 * 32; i += 256) {
        int r = i / 32, j = i % 32;
        float sa = 0.f, sb = 0.f;
#pragma unroll
        for (int k = 0; k < 5; k++) { sa += HA[(r + k) * 32 + j]; sb += HB[(r + k) * 32 + j]; }
        int gy = oy + r, gx = ox + j;
        float inv = 1.0f / (wcount(gy, H2, 2) * wcount(gx, W2, 2));
        float q = sa * inv * Tin[(r + 4) * 42 + (j + 4)] + sb * inv;
        PDN[(size_t)plane * H2 * W2 + (size_t)gy * W2 + gx] = q;
        lsum += q;
    }
    // deterministic block reduction (fixed tree order)
#pragma unroll
    for (int off = 16; off > 0; off >>= 1) lsum += __shfl_down(lsum, off, 32);
    if ((tid & 31) == 0) red[tid >> 5] = lsum;
    __syncthreads();
    if (tid == 0) {
        float s = 0.f;
#pragma unroll
        for (int w = 0; w < 8; w++) s += red[w];
        BSUM[plane * (NBLK_X * NBLK_Y) + blockIdx.y * NBLK_X + blockIdx.x] = s;
    }
}

// =============== K3: AWB gains (deterministic reduction of block partials) ===============
__global__ __launch_bounds__(256) void k_gains(const float* __restrict__ BSUM,
                                               float* __restrict__ gains) {
    __shared__ float sh[256];
    __shared__ float tot[4];
    const int NB = NBLK_X * NBLK_Y;
    for (int p = 0; p < 4; p++) {
        float s = 0.f;
        for (int i = threadIdx.x; i < NB; i += 256) s += BSUM[p * NB + i];
        sh[threadIdx.x] = s;
        __syncthreads();
        for (int o = 128; o > 0; o >>= 1) {
            if (threadIdx.x < o) sh[threadIdx.x] += sh[threadIdx.x + o];
            __syncthreads();
        }
        if (threadIdx.x == 0) tot[p] = sh[0];
        __syncthreads();
    }
    if (threadIdx.x == 0) {
        const float N = (float)(H2 * W2);
        float r = tot[0] / N, gm = 0.5f * (tot[1] / N + tot[2] / N), b = tot[3] / N;
        gains[0] = fminf(gm / (r + 1e-8f), 4.0f);
        gains[1] = fminf(gm / (b + 1e-8f), 4.0f);
    }
}

// =============== K4: AWB apply + demosaic + CCM + luma + 2x2 Ys ===============
__global__ __launch_bounds__(256) void k_demosaic(const float* __restrict__ PDN,
                                                  const float* __restrict__ ccm,
                                                  const float* __restrict__ gains,
                                                  float* __restrict__ RGB,
                                                  float* __restrict__ Ys) {
    const int oy = blockIdx.y * 32, ox = blockIdx.x * 32;
    const int tid = threadIdx.x;
    __shared__ float Tm[34 * 36];

    const float gr = gains[0], gb = gains[1];
    // build gained+clipped mosaic tile (34x34, halo 1, zero pad = conv 'SAME')
    for (int i = tid; i < 34 * 34; i += 256) {
        int r = i / 34, c = i % 34;
        int gi = oy - 1 + r, gj = ox - 1 + c;
        float v = 0.0f;
        if (gi >= 0 && gi < HH && gj >= 0 && gj < WW) {
            int p = 2 * (gi & 1) + (gj & 1);
            v = PDN[(size_t)p * H2 * W2 + (size_t)(gi >> 1) * W2 + (gj >> 1)];
            if (p == 0) v *= gr;
            if (p == 3) v *= gb;
            v = clip01(v);
        }
        Tm[r * 36 + c] = v;
    }
    __syncthreads();

    const float c00 = ccm[0], c01 = ccm[1], c02 = ccm[2];
    const float c10 = ccm[3], c11 = ccm[4], c12 = ccm[5];
    const float c20 = ccm[6], c21 = ccm[7], c22 = ccm[8];
    const float krb[3][3] = {{0.25f, 0.5f, 0.25f}, {0.5f, 1.0f, 0.5f}, {0.25f, 0.5f, 0.25f}};
    const float kg[3][3]  = {{0.0f, 0.25f, 0.0f}, {0.25f, 1.0f, 0.25f}, {0.0f, 0.25f, 0.0f}};

    const int ty = tid >> 4, tx = tid & 15;   // each thread: one 2x2 pixel block
    float sumY = 0.0f;
    float* Rp = RGB;
    float* Gp = RGB + (size_t)HH * WW;
    float* Bp = RGB + 2 * (size_t)HH * WW;
    for (int dy = 0; dy < 2; dy++) {
        float2 rv, gv, bv;
        for (int dx = 0; dx < 2; dx++) {
            int i = oy + 2 * ty + dy, j = ox + 2 * tx + dx;
            int r = 2 * ty + dy + 1, c = 2 * tx + dx + 1;
            float R = 0.f, G = 0.f, B = 0.f;
#pragma unroll
            for (int di = -1; di <= 1; di++)
#pragma unroll
                for (int dj = -1; dj <= 1; dj++) {
                    float v = Tm[(r + di) * 36 + (c + dj)];
                    int pi = (i + di) & 1, pj = (j + dj) & 1;
                    if (pi == 0 && pj == 0) R += v * krb[di + 1][dj + 1];
                    else if (pi == 1 && pj == 1) B += v * krb[di + 1][dj + 1];
                    else G += v * kg[di + 1][dj + 1];
                }
            float r2 = clip01(c00 * R + c01 * G + c02 * B);
            float g2 = clip01(c10 * R + c11 * G + c12 * B);
            float b2 = clip01(c20 * R + c21 * G + c22 * B);
            sumY += 0.299f * r2 + 0.587f * g2 + 0.114f * b2;
            if (dx == 0) { rv.x = r2; gv.x = g2; bv.x = b2; }
            else         { rv.y = r2; gv.y = g2; bv.y = b2; }
        }
        size_t row = (size_t)(oy + 2 * ty + dy) * WW + ox;
        ((float2*)(Rp + row))[tx] = rv;
        ((float2*)(Gp + row))[tx] = gv;
        ((float2*)(Bp + row))[tx] = bv;
    }
    Ys[(size_t)(oy / 2 + ty) * W2 + (ox / 2 + tx)] = sumY * 0.25f;
}

// =============== K5: fused LTM guided filter (r=8, eps=1e-3) on Ys ===============
// 17-tap horizontal box sums are banded matmuls on the WMMA pipe:
//   Out(16x16) = A(16 rows x 4 cols) * B(4x16 0/1 band), accumulated over 8 K-chunks.
#define LEPS 1e-3f
__global__ __launch_bounds__(256) void k_ltm(const float* __restrict__ Ys,
                                             float* __restrict__ BS) {
    const int oy = blockIdx.y * 32, ox = blockIdx.x * 32;
    const int tid = threadIdx.x;
    const int wave = tid >> 5, lane = tid & 31;
    const int lm = lane & 15, hv = lane >> 4;

    __shared__ float smem[13824];          // 54 KB, regions aliased below
    float* Tin = smem;                     // 64 x pitch 66 (16.5 KB)  [dies after step1]
    float* Aa  = smem;                     // 48x48 a  (aliases Tin)
    float* Bb  = smem + 48 * 48;           // 48x48 b
    float* Hs  = smem + 4608;              // 64x48 horiz sums of I
    float* Hs2 = Hs + 64 * 48;             // 64x48 horiz sums of I^2
    float* HA  = Hs2 + 64 * 48;            // 48x32 horiz sums of a
    float* HB  = HA + 48 * 32;             // 48x32 horiz sums of b

    tile_load(Tin, 66, Ys, oy - 16, ox - 16, 64, 64, H2, W2, tid, 256);
    __syncthreads();

    // stash the 32x32 center I values before Aa/Bb overwrite Tin
    float cI[4];
#pragma unroll
    for (int u = 0; u < 4; u++) {
        int idx = tid + 256 * u;
        cI[u] = Tin[(16 + (idx >> 5)) * 66 + 16 + (idx & 31)];
    }

    // ---- step1: horizontal 17-tap sums of I and I^2 (64 rows x 48 cols) ----
#if USE_WMMA
    for (int job = wave; job < 24; job += 8) {   // wave-uniform: EXEC stays full
        int tile = job >> 1, arr = job & 1;
        int R0 = (tile / 3) * 16, C0 = (tile % 3) * 16;
        v8f acc = {0.f, 0.f, 0.f, 0.f, 0.f, 0.f, 0.f, 0.f};
#pragma unroll
        for (int t = 0; t < 8; t++) {
            int colA = C0 + 4 * t + 2 * hv;
            float ax = Tin[(R0 + lm) * 66 + colA];
            float ay = Tin[(R0 + lm) * 66 + colA + 1];
            if (arr) { ax *= ax; ay *= ay; }
            int kx = 4 * t + 2 * hv;                  // K index of B.x ; B.y = kx+1
            float bx = (lm >= kx - 16 && lm <= kx) ? 1.0f : 0.0f;
            float by = (lm >= kx - 15 && lm <= kx + 1) ? 1.0f : 0.0f;
            v2f A = {ax, ay};
            v2f B = {bx, by};
            acc = __builtin_amdgcn_wmma_f32_16x16x4_f32(false, A, false, B, (short)0, acc,
                                                        false, false);
        }
        float* dst = arr ? Hs2 : Hs;
#pragma unroll
        for (int v = 0; v < 8; v++) dst[(R0 + v + 8 * hv) * 48 + C0 + lm] = acc[v];
    }
#else
    for (int i = tid; i < 64 * 48; i += 256) {
        int r = i / 48, j = i % 48;
        float s = 0.f, s2 = 0.f;
        for (int t = 0; t < 17; t++) { float v = Tin[r * 66 + j + t]; s += v; s2 += v * v; }
        Hs[i] = s; Hs2[i] = s2;
    }
#endif
    __syncthreads();

    // ---- step2: vertical 17-tap -> m, var -> a, b (48x48) ----
    for (int i = tid; i < 48 * 48; i += 256) {
        int r = i / 48, j = i % 48;
        float s = 0.f, s2 = 0.f;
        for (int t = 0; t < 17; t++) { s += Hs[(r + t) * 48 + j]; s2 += Hs2[(r + t) * 48 + j]; }
        int gy = oy - 8 + r, gx = ox - 8 + j;
        float inv = 1.0f / (wcount(gy, H2, 8) * wcount(gx, W2, 8));
        float m = s * inv, m2 = s2 * inv;
        float var = m2 - m * m;
        float a = var / (var + LEPS);
        Aa[i] = a; Bb[i] = m * (1.0f - a);
    }
    __syncthreads();

    // ---- step3: horizontal 17-tap of a, b (48 rows x 32 cols), WMMA again ----
#if USE_WMMA
    for (int job = wave; job < 12; job += 8) {
        int tile = job >> 1, arr = job & 1;
        int R0 = (tile >> 1) * 16, C0 = (tile & 1) * 16;
        const float* src = arr ? Bb : Aa;
        v8f acc = {0.f, 0.f, 0.f, 0.f, 0.f, 0.f, 0.f, 0.f};
#pragma unroll
        for (int t = 0; t < 8; t++) {
            int colA = C0 + 4 * t + 2 * hv;
            float ax = src[(R0 + lm) * 48 + colA];
            float ay = src[(R0 + lm) * 48 + colA + 1];
            int kx = 4 * t + 2 * hv;
            float bx = (lm >= kx - 16 && lm <= kx) ? 1.0f : 0.0f;
            float by = (lm >= kx - 15 && lm <= kx + 1) ? 1.0f : 0.0f;
            v2f A = {ax, ay};
            v2f B = {bx, by};
            acc = __builtin_amdgcn_wmma_f32_16x16x4_f32(false, A, false, B, (short)0, acc,
                                                        false, false);
        }
        float* dst = arr ? HB : HA;
#pragma unroll
        for (int v = 0; v < 8; v++) dst[(R0 + v + 8 * hv) * 32 + C0 + lm] = acc[v];
    }
#else
    for (int i = tid; i < 48 * 32; i += 256) {
        int r = i / 32, j = i % 32;
        float sa = 0.f, sb = 0.f;
        for (int t = 0; t < 17; t++) { sa += Aa[r * 48 + j + t]; sb += Bb[r * 48 + j + t]; }
        HA[i] = sa; HB[i] = sb;
    }
#endif
    __syncthreads();

    // ---- step4: vertical 17-tap -> base = mean(a)*I + mean(b) ----
#pragma unroll
    for (int u = 0; u < 4; u++) {
        int idx = tid + 256 * u;
        int r = idx >> 5, j = idx & 31;
        float sa = 0.f, sb = 0.f;
        for (int t = 0; t < 17; t++) { sa += HA[(r + t) * 32 + j]; sb += HB[(r + t) * 32 + j]; }
        int gy = oy + r, gx = ox + j;
        float inv = 1.0f / (wcount(gy, H2, 8) * wcount(gx, W2, 8));
        BS[(size_t)gy * W2 + gx] = sa * inv * cI[u] + sb * inv;
    }
}

// =============== K6: upsample base + tone map + gamma + RGB->NV12 ===============
__device__ __forceinline__ float sampleBS(const float* __restrict__ BS, float sy, float sx) {
    sy = fminf(fmaxf(sy, 0.0f), (float)(H2 - 1));
    sx = fminf(fmaxf(sx, 0.0f), (float)(W2 - 1));
    int y0 = (int)sy, x0 = (int)sx;
    int y1 = min(y0 + 1, H2 - 1), x1 = min(x0 + 1, W2 - 1);
    float fy = sy - (float)y0, fx = sx - (float)x0;
    float v00 = BS[(size_t)y0 * W2 + x0], v01 = BS[(size_t)y0 * W2 + x1];
    float v10 = BS[(size_t)y1 * W2 + x0], v11 = BS[(size_t)y1 * W2 + x1];
    return (v00 * (1.f - fx) + v01 * fx) * (1.f - fy) + (v10 * (1.f - fx) + v11 * fx) * fy;
}

__global__ __launch_bounds__(256) void k_finish(const float* __restrict__ RGB,
                                                const float* __restrict__ BS,
                                                float* __restrict__ out) {
    int idx = blockIdx.x * 256 + threadIdx.x;    // one thread per 2x2 output block
    int y2 = idx / W2, x2 = idx % W2;
    const float* Rp = RGB;
    const float* Gp = RGB + (size_t)HH * WW;
    const float* Bp = RGB + 2 * (size_t)HH * WW;

    float Us = 0.f, Vs = 0.f;
    for (int dy = 0; dy < 2; dy++) {
        int i = 2 * y2 + dy;
        size_t row = (size_t)i * WW;
        float2 rv = ((const float2*)(Rp + row))[x2];
        float2 gv = ((const float2*)(Gp + row))[x2];
        float2 bv = ((const float2*)(Bp + row))[x2];
        for (int dx = 0; dx < 2; dx++) {
            int j = 2 * x2 + dx;
            float r = dx ? rv.y : rv.x, g = dx ? gv.y : gv.x, b = dx ? bv.y : bv.x;
            float Y = 0.299f * r + 0.587f * g + 0.114f * b;
            float base = sampleBS(BS, (i + 0.5f) * 0.5f - 0.5f, (j + 0.5f) * 0.5f - 0.5f);
            float Ynew = Y - 0.3f * base;                       // 0.7*base + (Y-base)
            float s = Ynew / (Y + 1e-6f);
            r = clip01(r * s); g = clip01(g * s); b = clip01(b * s);
            r = powf(fmaxf(r, 1e-6f), 1.0f / 2.2f);
            g = powf(fmaxf(g, 1e-6f), 1.0f / 2.2f);
            b = powf(fmaxf(b, 1e-6f), 1.0f / 2.2f);
            float Yf = 0.299f * r + 0.587f * g + 0.114f * b;
            Us += -0.168736f * r - 0.331264f * g + 0.5f * b + 0.5f;
            Vs += 0.5f * r - 0.418688f * g - 0.081312f * b + 0.5f;
            out[row + j] = fminf(fmaxf(Yf * 255.0f, 0.0f), 255.0f);
        }
    }
    size_t uv = (size_t)HH * WW + (size_t)y2 * WW + 2 * x2;
    out[uv]     = fminf(fmaxf(Us * 0.25f * 255.0f, 0.0f), 255.0f);
    out[uv + 1] = fminf(fmaxf(Vs * 0.25f * 255.0f, 0.0f), 255.0f);
}

// =============== host-side launcher ===============
extern "C" void kernel_launch(void* const* d_in, const int* in_sizes, int n_in,
                              void* d_out, int out_size, void* d_ws, size_t ws_size,
                              hipStream_t stream) {
    const float* x = (const float*)d_in[0];
    const float* ccm = (const float*)d_in[1];
    float* ws = (float*)d_ws;

    // workspace layout (floats); P aliases the low 1/3 of RGB (dead before RGB is written)
    float* RGB  = ws;                                  // 3*H*W   = 37,748,736
    float* P    = ws;                                  // 4*H2*W2 = 12,582,912 (alias)
    float* PDN  = RGB + (size_t)3 * HH * WW;           // 4*H2*W2
    float* Ysb  = PDN + (size_t)4 * H2 * W2;           // H2*W2
    float* BS   = Ysb + (size_t)H2 * W2;               // H2*W2
    float* BSUM = BS + (size_t)H2 * W2;                // 4*3072
    float* GNS  = BSUM + 4 * NBLK_X * NBLK_Y;          // 2
    float* out  = (float*)d_out;

    k_decompand<<<(H2 * W2) / 256, 256, 0, stream>>>(x, P);
    k_denoise<<<dim3(NBLK_X, NBLK_Y, 4), 256, 0, stream>>>(P, PDN, BSUM);
    k_gains<<<1, 256, 0, stream>>>(BSUM, GNS);
    k_demosaic<<<dim3(WW / 32, HH / 32), 256, 0, stream>>>(PDN, ccm, GNS, RGB, Ysb);
    k_ltm<<<dim3(W2 / 32, H2 / 32), 256, 0, stream>>>(Ysb, BS);
    k_finish<<<(H2 * W2) / 256, 256, 0, stream>>>(RGB, BS, out);
}